// RelationalNetwork_30554397344541
// MI455X (gfx1250) — compile-verified
//
#include <hip/hip_runtime.h>

// RelationalNetwork forward for MI455X (gfx1250, wave32).
//
// Math simplification used: z[b,o,i,j] = A[b,o,j] + B[b,o,i] + b1[o] is a
// rank-1 outer sum, so instance-norm over (i,j) has separable statistics:
//   mean = meanA + meanB + b1  (b1 cancels after centering)
//   var  = varA + varB         (cross term is zero)
// => never materialize the 67 MB z tensor. s[i] = sum_j relu(a'[j] + b'[i]).
//
// GEMMs use V_WMMA_F32_16X16X4_F32 (full fp32 precision; GEMM flops are
// trivial here so low-precision WMMA buys nothing on the roofline).

typedef __attribute__((ext_vector_type(2))) float v2f;
typedef __attribute__((ext_vector_type(8))) float v8f;

static constexpr int kB   = 2;     // batch
static constexpr int kC   = 128;   // channels (c == d == o)
static constexpr int kHW  = 256;   // h*w
static constexpr int kW1LD = 384;  // w1 row stride (2c + d)
static constexpr int kCHW = kC * kHW;  // 32768
static constexpr float kEps = 1e-5f;

// ---- block-wide sum over 256 threads (8 wave32's) ----
__device__ __forceinline__ float blk_sum_256(float v, float* red) {
#pragma unroll
  for (int off = 16; off > 0; off >>= 1) v += __shfl_xor(v, off, 32);
  __syncthreads();  // protect red[] from any previous use
  if ((threadIdx.x & 31) == 0) red[threadIdx.x >> 5] = v;
  __syncthreads();
  float r = 0.f;
#pragma unroll
  for (int i = 0; i < 8; ++i) r += red[i];
  return r;
}

// ---- Stage 1: instance-norm + relu for x (+pos_emb) and y -----------------
// grid = 512 blocks: [0,256) -> x channels, [256,512) -> y channels.
__global__ __launch_bounds__(256) void in_norm_kernel(
    const float* __restrict__ x, const float* __restrict__ y,
    const float* __restrict__ pos, float* __restrict__ xn,
    float* __restrict__ yn) {
  __shared__ float red[8];
  const int tid = threadIdx.x;            // spatial position p in [0,256)
  const int id  = blockIdx.x;
  float v;
  float* dst;
  if (id < kB * kC) {                     // x path: id = b*128 + ch
    const int ch = id & (kC - 1);
    v = x[id * kHW + tid] + pos[tid * kC + ch];   // pos_emb[p, ch]
    dst = xn + id * kHW;
  } else {                                // y path
    const int id2 = id - kB * kC;
    v = y[id2 * kHW + tid];
    dst = yn + id2 * kHW;
  }
  const float s  = blk_sum_256(v, red);
  const float s2 = blk_sum_256(v * v, red);
  const float m   = s * (1.f / 256.f);
  const float var = s2 * (1.f / 256.f) - m * m;
  dst[tid] = fmaxf((v - m) * rsqrtf(var + kEps), 0.f);
}

// ---- Stage 2: A = Wi*X ; B = Wj*X + Wy*Y (per batch), fp32 WMMA -----------
// One wave per 16x16 output tile. 256 tiles total -> 64 blocks x 4 waves.
__global__ __launch_bounds__(128) void gemm_ab_kernel(
    const float* __restrict__ w1, const float* __restrict__ xn,
    const float* __restrict__ yn, float* __restrict__ Abuf,
    float* __restrict__ Bbuf) {
  const int lane = threadIdx.x & 31;
  const int wid  = threadIdx.x >> 5;
  const int wg   = blockIdx.x * 4 + wid;   // 0..255
  const int b    = wg >> 7;                // batch
  const int t    = wg & 127;
  const int mt   = t >> 4;                 // o-tile   0..7
  const int nt   = t & 15;                 // hw-tile  0..15

  // A-operand layout (16x4 f32): lane L<16 holds row L, K pair {k,k+1};
  // lane L>=16 holds row L-16, K pair {k+2,k+3}.
  const int arow = mt * 16 + (lane & 15);  // w1 row (o index)
  const int kq   = (lane >> 4) * 2;        // 0 or 2
  const int bcol = nt * 16 + (lane & 15);  // activation column (hw index)

  const float* Wi = w1;                    // w1[o*384 + c]
  const float* Wj = w1 + 128;
  const float* Wy = w1 + 256;
  const float* X  = xn + b * kCHW;
  const float* Y  = yn + b * kCHW;

  v8f accA = {};
  v8f accB = {};
  for (int kk = 0; kk < kC; kk += 4) {
    const int kb = kk + kq;
    v2f wa, wj, wy, xv, yv;
    wa.x = Wi[arow * kW1LD + kb];  wa.y = Wi[arow * kW1LD + kb + 1];
    wj.x = Wj[arow * kW1LD + kb];  wj.y = Wj[arow * kW1LD + kb + 1];
    wy.x = Wy[arow * kW1LD + kb];  wy.y = Wy[arow * kW1LD + kb + 1];
    xv.x = X[kb * kHW + bcol];     xv.y = X[(kb + 1) * kHW + bcol];
    yv.x = Y[kb * kHW + bcol];     yv.y = Y[(kb + 1) * kHW + bcol];
    accA = __builtin_amdgcn_wmma_f32_16x16x4_f32(false, wa, false, xv,
                                                 (short)0, accA, false, false);
    accB = __builtin_amdgcn_wmma_f32_16x16x4_f32(false, wj, false, xv,
                                                 (short)0, accB, false, false);
    accB = __builtin_amdgcn_wmma_f32_16x16x4_f32(false, wy, false, yv,
                                                 (short)0, accB, false, false);
  }

  // C/D layout: VGPR v -> row (v) for lanes 0-15, row (v+8) for lanes 16-31.
  const int srow = mt * 16 + (lane >> 4) * 8;
  const int scol = nt * 16 + (lane & 15);
  float* Ao = Abuf + b * kCHW;
  float* Bo = Bbuf + b * kCHW;
#pragma unroll
  for (int v = 0; v < 8; ++v) {
    Ao[(srow + v) * kHW + scol] = accA[v];
    Bo[(srow + v) * kHW + scol] = accB[v];
  }
}

// ---- Stage 3: separable instance-norm of outer sum + relu + sum_j --------
// One block per (b,o): stats of A-row and B-row, then s[i]=sum_j relu(a+b).
__global__ __launch_bounds__(256) void relsum_kernel(
    const float* __restrict__ Abuf, const float* __restrict__ Bbuf,
    float* __restrict__ Sbuf) {
  __shared__ float la[256];
  __shared__ float red[8];
  const int tid  = threadIdx.x;
  const int base = blockIdx.x * kHW;       // blockIdx = b*128 + o
  const float a  = Abuf[base + tid];
  const float bb = Bbuf[base + tid];
  const float sA  = blk_sum_256(a, red);
  const float sA2 = blk_sum_256(a * a, red);
  const float sB  = blk_sum_256(bb, red);
  const float sB2 = blk_sum_256(bb * bb, red);
  const float mA = sA * (1.f / 256.f), mB = sB * (1.f / 256.f);
  const float vA = sA2 * (1.f / 256.f) - mA * mA;
  const float vB = sB2 * (1.f / 256.f) - mB * mB;
  const float rsig = rsqrtf(vA + vB + kEps);   // b1 cancels after centering
  la[tid] = (a - mA) * rsig;
  const float bn = (bb - mB) * rsig;
  __syncthreads();
  float s = 0.f;
#pragma unroll 8
  for (int j = 0; j < 256; ++j) s += fmaxf(la[j] + bn, 0.f);  // LDS broadcast
  Sbuf[base + tid] = s;
}

// ---- Stage 4: out = w2 * S + 256*b2, fp32 WMMA ---------------------------
__global__ __launch_bounds__(128) void gemm_out_kernel(
    const float* __restrict__ w2, const float* __restrict__ b2,
    const float* __restrict__ Sbuf, float* __restrict__ out) {
  const int lane = threadIdx.x & 31;
  const int wid  = threadIdx.x >> 5;
  const int wg   = blockIdx.x * 4 + wid;
  const int b    = wg >> 7;
  const int t    = wg & 127;
  const int mt   = t >> 4;
  const int nt   = t & 15;

  const int prow = mt * 16 + (lane & 15);  // w2 row (p index)
  const int kq   = (lane >> 4) * 2;
  const int bcol = nt * 16 + (lane & 15);

  const float* S = Sbuf + b * kCHW;
  v8f acc = {};
  for (int kk = 0; kk < kC; kk += 4) {
    const int kb = kk + kq;
    v2f wv, sv;
    wv.x = w2[prow * kC + kb];   wv.y = w2[prow * kC + kb + 1];
    sv.x = S[kb * kHW + bcol];   sv.y = S[(kb + 1) * kHW + bcol];
    acc = __builtin_amdgcn_wmma_f32_16x16x4_f32(false, wv, false, sv,
                                                (short)0, acc, false, false);
  }

  const int srow = mt * 16 + (lane >> 4) * 8;
  const int scol = nt * 16 + (lane & 15);
  float* O = out + b * kCHW;
#pragma unroll
  for (int v = 0; v < 8; ++v)
    O[(srow + v) * kHW + scol] = acc[v] + 256.f * b2[srow + v];
}

extern "C" void kernel_launch(void* const* d_in, const int* in_sizes, int n_in,
                              void* d_out, int out_size, void* d_ws,
                              size_t ws_size, hipStream_t stream) {
  (void)in_sizes; (void)n_in; (void)out_size; (void)ws_size;
  const float* x   = (const float*)d_in[0];
  const float* y   = (const float*)d_in[1];
  const float* pos = (const float*)d_in[2];
  const float* w1  = (const float*)d_in[3];
  // d_in[4] = b1: cancels analytically inside the z instance-norm.
  const float* w2  = (const float*)d_in[5];
  const float* b2  = (const float*)d_in[6];
  float* out = (float*)d_out;

  float* xn   = (float*)d_ws;          // 65536 floats
  float* yn   = xn + kB * kCHW / 1;    // 65536
  float* Abuf = yn + kB * kCHW;        // 65536
  float* Bbuf = Abuf + kB * kCHW;      // 65536
  float* Sbuf = Bbuf + kB * kCHW;      // 65536  (total 1.25 MB of d_ws)

  in_norm_kernel<<<2 * kB * kC, 256, 0, stream>>>(x, y, pos, xn, yn);
  gemm_ab_kernel<<<64, 128, 0, stream>>>(w1, xn, yn, Abuf, Bbuf);
  relsum_kernel<<<kB * kC, 256, 0, stream>>>(Abuf, Bbuf, Sbuf);
  gemm_out_kernel<<<64, 128, 0, stream>>>(w2, b2, Sbuf, out);
}